// Sparsify_hypercol_74775380623609
// MI455X (gfx1250) — compile-verified
//
#include <hip/hip_runtime.h>

// Problem constants (fixed by the reference): x is [16, 128, 128, 128] f32.
#define N_   16
#define C_   128
#define HW_  16384          // 128*128
#define KTOP 1638           // int(0.1 * 16384)

typedef __attribute__((ext_vector_type(2))) float v2f;
typedef __attribute__((ext_vector_type(8))) float v8f;

// Monotonic float -> uint mapping (larger float => larger uint).
__device__ __forceinline__ unsigned fmap(float f) {
    unsigned u = __float_as_uint(f);
    return (u & 0x80000000u) ? ~u : (u | 0x80000000u);
}

// ---------------------------------------------------------------------------
// Kernel 1: per-column channel mean via V_WMMA_F32_16X16X4_F32.
// Each wave computes 16 column means: D = sum_k A(1/128) * B(x tile).
//   A: 16x4, all elements = 1/128  -> every row of D equals the column mean.
//   B: 4x16 tile, B[k][ncol] = x[n][c0+k][p0+ncol].
//   B VGPR layout (wave32): VGPR v, lane l: K = v + 2*(l>=16), N = l%16.
// 32 accumulating WMMAs cover the 128 channels. Row M=0 of D lives in
// acc[0] of lanes 0..15 -> those lanes store the 16 stats.
// ---------------------------------------------------------------------------
__global__ void colstats_wmma_kernel(const float* __restrict__ x,
                                     float* __restrict__ stats) {
    const int lane = threadIdx.x & 31;
    const int wid  = (blockIdx.x * (blockDim.x >> 5)) + (threadIdx.x >> 5);
    const long g0  = (long)wid * 16;          // flattened [n*HW] tile start
    const int n    = (int)(g0 >> 14);         // / HW_
    const int p0   = (int)(g0 & (HW_ - 1));
    const int hi   = lane >> 4;               // which half-wave
    const int col  = lane & 15;               // N index

    const float* base = x + ((size_t)n * C_) * HW_ + p0 + col;

#if __has_builtin(__builtin_amdgcn_wmma_f32_16x16x4_f32)
    v8f acc = {};
    v2f a;  a[0] = 1.0f / 128.0f;  a[1] = 1.0f / 128.0f;
    #pragma unroll 4
    for (int c0 = 0; c0 < C_; c0 += 4) {
        const float* bp = base + (size_t)(c0 + 2 * hi) * HW_;
        v2f b;  b[0] = bp[0];  b[1] = bp[HW_];
        acc = __builtin_amdgcn_wmma_f32_16x16x4_f32(
            /*neg_a=*/false, a, /*neg_b=*/false, b,
            /*c_mod=*/(short)0, acc, /*reuse_a=*/false, /*reuse_b=*/false);
    }
    if (lane < 16)
        stats[(size_t)n * HW_ + p0 + lane] = acc[0];
#else
    // Fallback (should not trigger on gfx1250): scalar channel mean.
    if (lane < 16) {
        float s = 0.f;
        for (int c = 0; c < C_; ++c) s += base[(size_t)c * HW_];
        stats[(size_t)n * HW_ + p0 + lane] = s * (1.0f / 128.0f);
    }
#endif
}

// ---------------------------------------------------------------------------
// Kernel 2: exact k-th largest per sample via 4-round radix select (8 bit
// digits, MSB first) on the uint-mapped stats. One workgroup per n.
// Outputs: thr[n] = mapped threshold value, needEq[n] = #ties to keep.
// ---------------------------------------------------------------------------
__global__ void radix_select_kernel(const float* __restrict__ stats,
                                    unsigned* __restrict__ thr,
                                    int* __restrict__ needEq) {
    const int n = blockIdx.x;
    const int T = blockDim.x;          // 256
    __shared__ unsigned hist[256];
    __shared__ unsigned s_prefix;
    __shared__ unsigned s_remk;
    __shared__ unsigned s_gt;

    if (threadIdx.x == 0) { s_prefix = 0u; s_remk = KTOP; s_gt = 0u; }
    __syncthreads();

    const float* sp = stats + (size_t)n * HW_;

    for (int shift = 24; shift >= 0; shift -= 8) {
        for (int i = threadIdx.x; i < 256; i += T) hist[i] = 0u;
        __syncthreads();
        const unsigned pmask  = (shift == 24) ? 0u : (0xFFFFFFFFu << (shift + 8));
        const unsigned prefix = s_prefix;
        for (int i = threadIdx.x; i < HW_; i += T) {
            unsigned u = fmap(sp[i]);
            if ((u & pmask) == prefix)
                atomicAdd(&hist[(u >> shift) & 0xFFu], 1u);
        }
        __syncthreads();
        if (threadIdx.x == 0) {
            unsigned k = s_remk, cum = 0u;
            int b = 255;
            for (; b > 0; --b) {
                if (cum + hist[b] >= k) break;
                cum += hist[b];
            }
            s_prefix = prefix | ((unsigned)b << shift);
            s_remk   = k - cum;
        }
        __syncthreads();
    }

    const unsigned ut = s_prefix;
    unsigned loc = 0u;
    for (int i = threadIdx.x; i < HW_; i += T)
        if (fmap(sp[i]) > ut) loc++;
    atomicAdd(&s_gt, loc);
    __syncthreads();
    if (threadIdx.x == 0) {
        thr[n]    = ut;
        needEq[n] = KTOP - (int)s_gt;  // >= 1; index-ordered ties to include
    }
}

// ---------------------------------------------------------------------------
// Kernel 3: streaming scale. out = x * (selected ? 1.0 : (1-tau)).
// Grid: (HW_/512, N_); 128 threads; thread owns 4 consecutive positions and
// streams all 128 channels with float4 (global_load/store_b128).
// x is L2-resident after kernel 1 (128 MiB < 192 MB L2).
// ---------------------------------------------------------------------------
__global__ void scale_kernel(const float* __restrict__ x,
                             const float* __restrict__ stats,
                             const unsigned* __restrict__ thr,
                             const int* __restrict__ needEq,
                             const float* __restrict__ tauPtr,
                             float* __restrict__ out) {
    const int n  = blockIdx.y;
    const int p0 = blockIdx.x * 512 + threadIdx.x * 4;
    const unsigned ut = thr[n];
    const int ne = needEq[n];
    const float lo = 1.0f - *tauPtr;
    const float* sp = stats + (size_t)n * HW_;

    float4 sc;
    float* scp = &sc.x;
    #pragma unroll
    for (int j = 0; j < 4; ++j) {
        const int p = p0 + j;
        const unsigned u = fmap(sp[p]);
        float s;
        if (u > ut) {
            s = 1.0f;
        } else if (u < ut) {
            s = lo;
        } else {
            // Rare: tie at the threshold. Index-ordered rank among equals
            // (matches top_k's lower-index-first tie break).
            int r = 0;
            for (int q = 0; q < p; ++q)
                if (fmap(sp[q]) == ut) r++;
            s = (r < ne) ? 1.0f : lo;
        }
        scp[j] = s;
    }

    const size_t nbase = (size_t)n * C_ * HW_ + p0;
    #pragma unroll 4
    for (int c = 0; c < C_; ++c) {
        const size_t off = nbase + (size_t)c * HW_;
        float4 v = *(const float4*)(x + off);
        v.x *= sc.x;  v.y *= sc.y;  v.z *= sc.z;  v.w *= sc.w;
        *(float4*)(out + off) = v;
    }
}

// ---------------------------------------------------------------------------
extern "C" void kernel_launch(void* const* d_in, const int* in_sizes, int n_in,
                              void* d_out, int out_size, void* d_ws, size_t ws_size,
                              hipStream_t stream) {
    (void)in_sizes; (void)n_in; (void)out_size; (void)ws_size;
    const float* x   = (const float*)d_in[0];
    const float* tau = (const float*)d_in[1];
    float* out       = (float*)d_out;

    // Workspace layout: [stats: N_*HW_ f32 = 1 MiB][thr: N_ u32][needEq: N_ i32]
    float*    stats  = (float*)d_ws;
    unsigned* thr    = (unsigned*)((char*)d_ws + (size_t)N_ * HW_ * sizeof(float));
    int*      needEq = (int*)((char*)d_ws + (size_t)N_ * HW_ * sizeof(float)
                                          + (size_t)N_ * sizeof(unsigned));

    // Kernel 1: 16384 waves total (one per 16-position tile), 8 waves/block.
    const int totalWaves = (N_ * HW_) / 16;            // 16384
    colstats_wmma_kernel<<<totalWaves / 8, 256, 0, stream>>>(x, stats);

    // Kernel 2: one workgroup per sample.
    radix_select_kernel<<<N_, 256, 0, stream>>>(stats, thr, needEq);

    // Kernel 3: streaming scale, float4 vectorized.
    dim3 grid(HW_ / 512, N_);
    scale_kernel<<<grid, 128, 0, stream>>>(x, stats, thr, needEq, tau, out);
}